// StructuralAttentionLayer_79156247265995
// MI455X (gfx1250) — compile-verified
//
#include <hip/hip_runtime.h>

typedef float v2f __attribute__((ext_vector_type(2)));
typedef float v8f __attribute__((ext_vector_type(8)));

#define FDIM 128   // feature dim D = H*C = 128

// ---------------------------------------------------------------------------
// Projection GEMM: Y = X @ W^T   (X: nrows x 128, W: 128 x 128 row-major)
// One wave computes one 16-row tile, all 128 output columns (8 col-tiles),
// via 32 k-steps of V_WMMA_F32_16X16X4_F32.
// Fragment layouts per CDNA5 ISA §7.12.2:
//   A 16x4 f32 : lane l -> M = l&15, VGPR v -> K = 2*(l>>4) + v   (contig pair)
//   B 4x16 f32 : lane l -> N = l&15, VGPR v -> K = 2*(l>>4) + v   (contig pair of W row n)
//   C 16x16 f32: lane l -> N = l&15, VGPR v -> M = v + 8*(l>>4)
// Tail rows are handled by CLAMPING the A row index (garbage in row m only
// corrupts output row m, which is never stored) -> EXEC stays all-1s, no
// predication in the inner loop.
// ---------------------------------------------------------------------------
__global__ __launch_bounds__(128)
void proj_wmma_kernel(const float* __restrict__ X,
                      const float* __restrict__ Wa,
                      const float* __restrict__ Wb,
                      const float* __restrict__ Wc,
                      float* __restrict__ Ya,
                      float* __restrict__ Yb,
                      float* __restrict__ Yc,
                      int nrows)
{
    const float* W = (blockIdx.y == 0) ? Wa : (blockIdx.y == 1) ? Wb : Wc;
    float*       Y = (blockIdx.y == 0) ? Ya : (blockIdx.y == 1) ? Yb : Yc;

    const int lane     = threadIdx.x & 31;
    const int wave     = threadIdx.x >> 5;
    const int row_base = (blockIdx.x * 4 + wave) * 16;
    if (row_base >= nrows) return;

    const int mn    = lane & 15;   // M (for A) / N (for B,C) within tile
    const int khalf = lane >> 4;   // k-pair selector

    int arow = row_base + mn;
    if (arow > nrows - 1) arow = nrows - 1;          // clamp, keep EXEC full
    const float* xrow = X + (size_t)arow * FDIM;

    v8f acc[8];
    v8f zero = {0.f, 0.f, 0.f, 0.f, 0.f, 0.f, 0.f, 0.f};
    #pragma unroll
    for (int nt = 0; nt < 8; ++nt) acc[nt] = zero;

    for (int kt = 0; kt < 32; ++kt) {
        const int kk = kt * 4 + khalf * 2;           // even -> 8B aligned
        v2f a = *(const v2f*)(xrow + kk);
        #pragma unroll
        for (int nt = 0; nt < 8; ++nt) {
            const float* wrow = W + (size_t)(nt * 16 + mn) * FDIM + kk;
            v2f b = *(const v2f*)wrow;               // B[k][n] = W[n][k]
            acc[nt] = __builtin_amdgcn_wmma_f32_16x16x4_f32(
                false, a, false, b, (short)0, acc[nt], false, false);
        }
    }

    #pragma unroll
    for (int nt = 0; nt < 8; ++nt) {
        const int col = nt * 16 + mn;
        #pragma unroll
        for (int v = 0; v < 8; ++v) {
            const int r = row_base + khalf * 8 + v;
            if (r < nrows) Y[(size_t)r * FDIM + col] = acc[nt][v];
        }
    }
}

// ---------------------------------------------------------------------------
// Edge-MLP attention logits via WMMA.
// Flattened rows m = e*8 + h (M = E*8), K = 48 (Hs|Ho|Hr segments of 16),
// N = 16 hidden units. One wave = one 16-row tile = 12 k-steps of
// V_WMMA_F32_16X16X4_F32 with gathered A fragments (one aligned b64 load per
// lane per step: k-pairs never straddle a 16-float segment boundary).
// Epilogue: LeakyReLU, scale by w2[n], 16-lane-half xor-shuffle reduction
// (row m's 16 columns live in one half-wave at VGPR m&7), lanes 0/16 emit the
// logit and the order-preserving-uint segment max.
// ---------------------------------------------------------------------------
__global__ __launch_bounds__(256)
void edge_logit_wmma_kernel(const float* __restrict__ Hs, const float* __restrict__ Ho,
                            const float* __restrict__ Hr,
                            const int* __restrict__ esrc, const int* __restrict__ edst,
                            const int* __restrict__ ert,
                            const float* __restrict__ w1W, const float* __restrict__ w1b,
                            const float* __restrict__ w2W,
                            float* __restrict__ blog, unsigned int* __restrict__ mkey,
                            int E, int R)
{
    const int Mtot  = E * 8;
    const int lane  = threadIdx.x & 31;
    const int wave  = threadIdx.x >> 5;
    const int base  = (blockIdx.x * 8 + wave) * 16;
    if (base >= Mtot) return;

    const int n     = lane & 15;       // N (hidden unit) for B/C; row-in-tile for A
    const int khalf = lane >> 4;

    // --- B fragments (w1^T) + bias + w2, all tiny & cache-hot ---
    v2f bfrag[12];
    #pragma unroll
    for (int kt = 0; kt < 12; ++kt) {
        const int k = kt * 4 + khalf * 2;            // even -> 8B aligned
        bfrag[kt] = *(const v2f*)(w1W + n * 48 + k); // B[k][n] = w1W[n][k]
    }
    const float bias = w1b[n];
    const float w2v  = w2W[n];

    // --- A row setup: this lane's row (clamped for the tail tile) ---
    int ro = base + n;
    if (ro > Mtot - 1) ro = Mtot - 1;
    const int e  = ro >> 3;
    const int hh = ro & 7;
    const int src = esrc[e];
    const int dst = edst[e];
    int rt = ert[e]; if (rt > R - 1) rt = R - 1;

    const float* ps = Hs + (size_t)src * FDIM + hh * 16 + khalf * 2;
    const float* po = Ho + (size_t)dst * FDIM + hh * 16 + khalf * 2;
    const float* pr = Hr + (size_t)rt  * FDIM + hh * 16 + khalf * 2;

    // Two accumulator chains for WMMA ILP; bias folded into chain 0.
    v8f acc0 = {bias, bias, bias, bias, bias, bias, bias, bias};
    v8f acc1 = {0.f, 0.f, 0.f, 0.f, 0.f, 0.f, 0.f, 0.f};

    #pragma unroll
    for (int kt = 0; kt < 12; ++kt) {
        const float* psel = (kt < 4) ? ps : (kt < 8) ? po : pr;
        v2f a = *(const v2f*)(psel + (kt & 3) * 4);
        if (kt & 1)
            acc1 = __builtin_amdgcn_wmma_f32_16x16x4_f32(
                false, a, false, bfrag[kt], (short)0, acc1, false, false);
        else
            acc0 = __builtin_amdgcn_wmma_f32_16x16x4_f32(
                false, a, false, bfrag[kt], (short)0, acc0, false, false);
    }

    // --- LeakyReLU, scale by w2[n], reduce over n within each 16-lane half ---
    v8f red;
    #pragma unroll
    for (int v = 0; v < 8; ++v) {
        const float cv = acc0[v] + acc1[v];
        const float lv = cv > 0.f ? cv : 0.01f * cv;
        red[v] = lv * w2v;
    }
    #pragma unroll
    for (int m = 1; m < 16; m <<= 1) {
        #pragma unroll
        for (int v = 0; v < 8; ++v)
            red[v] += __shfl_xor(red[v], m, 32);
    }

    // Lanes 0 / 16 hold rows base+0..7 / base+8..15 (vgpr v -> row offset v).
    if (n == 0) {
        #pragma unroll
        for (int v = 0; v < 8; ++v) {
            const int row = base + khalf * 8 + v;
            if (row < Mtot) {
                const float bv = red[v];
                blog[row] = bv;
                const int e2 = row >> 3;
                const int h2 = row & 7;
                const int s2 = esrc[e2];
                const unsigned ub  = __float_as_uint(bv);
                const unsigned key = (ub & 0x80000000u) ? ~ub : (ub | 0x80000000u);
                atomicMax(&mkey[(size_t)s2 * 8 + h2], key);
            }
        }
    }
}

// ---------------------------------------------------------------------------
// ex = exp(b - m[src]); segment-sum denominators via atomicAdd. In-place over blog.
// ---------------------------------------------------------------------------
__global__ __launch_bounds__(256)
void edge_exp_kernel(float* __restrict__ blog, const unsigned* __restrict__ mkey,
                     float* __restrict__ denom, const int* __restrict__ esrc, int E)
{
    const long long gid = (long long)blockIdx.x * 256 + threadIdx.x;
    if (gid >= (long long)E * 8) return;
    const int e  = (int)(gid >> 3);
    const int hh = (int)(gid & 7);
    const int src = esrc[e];
    const unsigned key = mkey[(size_t)src * 8 + hh];
    const unsigned ub  = (key & 0x80000000u) ? (key ^ 0x80000000u) : ~key;
    const float m = __uint_as_float(ub);
    const float exv = __expf(blog[gid] - m);
    blog[gid] = exv;
    atomicAdd(&denom[(size_t)src * 8 + hh], exv);
}

// ---------------------------------------------------------------------------
// msg = (Hs[src] + Hr[rt]) * alpha, scatter-add into output accumulator.
// One thread per (edge, channel).
// ---------------------------------------------------------------------------
__global__ __launch_bounds__(256)
void edge_msg_kernel(const float* __restrict__ Hs, const float* __restrict__ Hr,
                     const float* __restrict__ ex, const float* __restrict__ denom,
                     const int* __restrict__ esrc, const int* __restrict__ edst,
                     const int* __restrict__ ert, float* __restrict__ acc, int E, int R)
{
    const long long gid = (long long)blockIdx.x * 256 + threadIdx.x;
    if (gid >= (long long)E * FDIM) return;
    const int e  = (int)(gid >> 7);
    const int ch = (int)(gid & 127);
    const int hh = ch >> 4;
    const int src = esrc[e];
    const int dst = edst[e];
    int rt = ert[e]; if (rt > R - 1) rt = R - 1;
    const float alpha = ex[(size_t)e * 8 + hh] / denom[(size_t)src * 8 + hh];
    const float v = (Hs[(size_t)src * FDIM + ch] + Hr[(size_t)rt * FDIM + ch]) * alpha;
    atomicAdd(&acc[(size_t)dst * FDIM + ch], v);
}

// ---------------------------------------------------------------------------
// out = relu(acc + Hw); also emit arange(N) (second tuple output, int32).
// ---------------------------------------------------------------------------
__global__ __launch_bounds__(256)
void finalize_kernel(float* __restrict__ out, const float* __restrict__ Hw,
                     int* __restrict__ arange, int N)
{
    const long long gid = (long long)blockIdx.x * 256 + threadIdx.x;
    if (gid >= (long long)N * FDIM) return;
    const float v = out[gid] + Hw[gid];
    out[gid] = v > 0.f ? v : 0.f;
    if ((gid & 127) == 0) {
        const int n = (int)(gid >> 7);
        arange[n] = n;
    }
}

extern "C" void kernel_launch(void* const* d_in, const int* in_sizes, int n_in,
                              void* d_out, int out_size, void* d_ws, size_t ws_size,
                              hipStream_t stream)
{
    const float* h   = (const float*)d_in[0];
    const float* h0  = (const float*)d_in[1];
    const float* W_s = (const float*)d_in[2];
    const float* W_o = (const float*)d_in[3];
    const float* W_r = (const float*)d_in[4];
    const float* W_w = (const float*)d_in[5];
    const float* w1W = (const float*)d_in[6];
    const float* w1b = (const float*)d_in[7];
    const float* w2W = (const float*)d_in[8];
    const int* esrc  = (const int*)d_in[9];
    const int* edst  = (const int*)d_in[10];
    const int* ert   = (const int*)d_in[11];

    const int N = in_sizes[0] / FDIM;
    const int R = in_sizes[1] / FDIM;
    const int E = in_sizes[9];

    // Workspace layout (fp32 unless noted)
    float* Hs   = (float*)d_ws;
    float* Ho_  = Hs  + (size_t)N * FDIM;
    float* Hw   = Ho_ + (size_t)N * FDIM;
    float* Hr   = Hw  + (size_t)N * FDIM;
    float* blog = Hr  + (size_t)R * FDIM;                 // E*8 logits, reused for exp
    unsigned* mkey = (unsigned*)(blog + (size_t)E * 8);   // N*8 order-preserving max keys
    float* denom   = (float*)(mkey + (size_t)N * 8);      // N*8 softmax denominators

    float* outv = (float*)d_out;
    int* arange = (int*)(outv + (size_t)N * FDIM);

    // Zero accumulators (graph-capture-safe async memsets).
    hipMemsetAsync(d_out, 0, (size_t)N * FDIM * sizeof(float), stream);
    hipMemsetAsync(mkey, 0, (size_t)N * 8 * (sizeof(unsigned) + sizeof(float)), stream);

    // Node projections Hs/Ho/Hw (one WMMA launch, 3 weight matrices via gridDim.y).
    {
        dim3 grid((N + 63) / 64, 3);
        proj_wmma_kernel<<<grid, 128, 0, stream>>>(h, W_s, W_o, W_w, Hs, Ho_, Hw, N);
    }
    // Relation projection Hr.
    {
        dim3 grid((R + 63) / 64, 1);
        proj_wmma_kernel<<<grid, 128, 0, stream>>>(h0, W_r, W_r, W_r, Hr, Hr, Hr, R);
    }
    // Edge logits (WMMA) + segment max.
    {
        const int Mtot   = E * 8;
        const int tiles  = (Mtot + 15) / 16;              // one wave per 16-row tile
        const int blocks = (tiles + 7) / 8;               // 8 waves per block
        edge_logit_wmma_kernel<<<blocks, 256, 0, stream>>>(Hs, Ho_, Hr, esrc, edst, ert,
                                                           w1W, w1b, w2W, blog, mkey, E, R);
    }
    // exp + denominators.
    {
        const long long tot = (long long)E * 8;
        const int blocks = (int)((tot + 255) / 256);
        edge_exp_kernel<<<blocks, 256, 0, stream>>>(blog, mkey, denom, esrc, E);
    }
    // Weighted messages scatter-added to destinations.
    {
        const long long tot = (long long)E * FDIM;
        const int blocks = (int)((tot + 255) / 256);
        edge_msg_kernel<<<blocks, 256, 0, stream>>>(Hs, Hr, blog, denom, esrc, edst, ert,
                                                    outv, E, R);
    }
    // Residual + ReLU + arange output.
    {
        const long long tot = (long long)N * FDIM;
        const int blocks = (int)((tot + 255) / 256);
        finalize_kernel<<<blocks, 256, 0, stream>>>(outv, Hw, arange, N);
    }
}